// DistWeightNeighbourLoss_47124381171855
// MI455X (gfx1250) — compile-verified
//
#include <hip/hip_runtime.h>
#include <hip/hip_bf16.h>

#define NROWS 4096
#define DIMS  128
#define MM    4
#define NNEG  (NROWS - MM)   // 4092
#define KPOS  (MM - 1)       // 3
#define PANEL_ROWS 16
#define PANEL_BYTES (PANEL_ROWS * NROWS * sizeof(float))   // 256 KB < 320 KB WGP LDS
#define NWAVES 16            // 512 threads = 16 wave32s (LDS caps us at 1 WG/WGP anyway)
#define TILE_ILP 4           // concurrent WMMA chains per wave (forces 4 loads in flight)

typedef __attribute__((ext_vector_type(2))) float v2f;
typedef __attribute__((ext_vector_type(8))) float v8f;

// ---------------------------------------------------------------- utilities

__device__ __forceinline__ float softplusf(float x) {
    if (x > 20.0f)  return x;
    if (x < -20.0f) return expf(x);
    return log1pf(expf(x));
}

// Deterministic per-(row,col) Gumbel noise via splitmix64-style hash.
__device__ __forceinline__ float gumbelf(int row, int col) {
    unsigned long long x = (((unsigned long long)(unsigned)row) << 32) | (unsigned)col;
    x ^= 0x9E3779B97F4A7C15ULL ^ (42ULL * 0xBF58476D1CE4E5B9ULL);
    x *= 0xBF58476D1CE4E5B9ULL; x ^= x >> 27;
    x *= 0x94D049BB133111EBULL; x ^= x >> 31;
    float u = ((float)(x >> 40) + 0.5f) * 5.9604645e-8f;  // (0,1), 24-bit mantissa
    return -logf(-logf(u));
}

__device__ __forceinline__ void top3_insert(float s, float v, float* ts, float* tv) {
    if (s > ts[2]) {
        if (s > ts[0]) {
            ts[2] = ts[1]; tv[2] = tv[1];
            ts[1] = ts[0]; tv[1] = tv[0];
            ts[0] = s;     tv[0] = v;
        } else if (s > ts[1]) {
            ts[2] = ts[1]; tv[2] = tv[1];
            ts[1] = s;     tv[1] = v;
        } else {
            ts[2] = s;     tv[2] = v;
        }
    }
}

// ---------------------------------------------------------------- kernels

__global__ void init_accum_kernel(float* accum) {
    if (threadIdx.x < 4) accum[threadIdx.x] = 0.0f;
}

// Row squared norms: one wave32 per row, 8 rows per 256-thread block.
__global__ void sqnorm_kernel(const float* __restrict__ A, float* __restrict__ sq) {
    const int lane = threadIdx.x & 31;
    const int row  = blockIdx.x * 8 + (threadIdx.x >> 5);
    const float* r = A + (size_t)row * DIMS;
    float s = 0.0f;
#pragma unroll
    for (int k = lane; k < DIMS; k += 32) { float v = r[k]; s += v * v; }
#pragma unroll
    for (int off = 16; off > 0; off >>= 1) s += __shfl_down(s, off, 32);
    if (lane == 0) sq[row] = s;
}

// Fused panel kernel: one workgroup (16 wave32s) per 16-row panel.
//  Phase 1: WMMA f32 Gram tiles -> distances, kept entirely in 256 KB LDS panel.
//           4 independent accumulator chains per wave so 4 B-loads must stay in
//           flight simultaneously (breaks the compiler's single-buffer sinking);
//           A-fragments are loop-invariant and get hoisted into registers.
//  Phase 2: one wave32 per row: neg mean/var (sort-invariant), Gumbel top-3 via
//           shuffle argmax extraction, softplus losses, scalar atomics.
__global__ void __launch_bounds__(32 * NWAVES)
panel_loss_kernel(const float* __restrict__ A,
                  const float* __restrict__ sq,
                  const int* __restrict__ targets,
                  float* __restrict__ accum) {
    extern __shared__ float panel[];             // [16][4096] distances
    const int tm   = blockIdx.x;                 // panel index, 0..255
    const int lane = threadIdx.x & 31;
    const int wave = threadIdx.x >> 5;           // 0..NWAVES-1
    const int half = lane >> 4;
    const int l    = lane & 15;

    // A fragment pointer (ISA 7.12.2 layout: lanes 0-15: M=lane, K=k0+0/1;
    // lanes 16-31: M=lane-16, K=k0+2/3)
    const float* __restrict__ Arow = A + (size_t)(tm * PANEL_ROWS + l) * DIMS + half * 2;

    float sqm[8];
#pragma unroll
    for (int r = 0; r < 8; ++r) sqm[r] = sq[tm * PANEL_ROWS + r + half * 8];

    // ---- Phase 1: 256 column tiles; each wave does 4 outer iters x 4 concurrent tiles
    for (int it = 0; it < (NROWS / 16) / (NWAVES * TILE_ILP); ++it) {
        int tnq[TILE_ILP];
        const float* __restrict__ Brow[TILE_ILP];
#pragma unroll
        for (int g = 0; g < TILE_ILP; ++g) {
            tnq[g]  = wave + NWAVES * it + (NWAVES * ((NROWS / 16) / (NWAVES * TILE_ILP))) * g;
            Brow[g] = A + (size_t)(tnq[g] * 16 + l) * DIMS + half * 2;
        }

        v8f c[TILE_ILP];
#pragma unroll
        for (int g = 0; g < TILE_ILP; ++g) c[g] = (v8f){};

#pragma unroll
        for (int k0 = 0; k0 < DIMS; k0 += 4) {
            const v2f a = *(const v2f*)(Arow + k0);
#pragma unroll
            for (int g = 0; g < TILE_ILP; ++g) {
                v2f b = *(const v2f*)(Brow[g] + k0);
                c[g] = __builtin_amdgcn_wmma_f32_16x16x4_f32(
                        /*neg_a=*/false, a, /*neg_b=*/false, b,
                        /*c_mod=*/(short)0, c[g], /*reuse_a=*/false, /*reuse_b=*/false);
            }
        }

#pragma unroll
        for (int g = 0; g < TILE_ILP; ++g) {
            const int   n   = tnq[g] * 16 + l;
            const float sqn = sq[n];
#pragma unroll
            for (int r = 0; r < 8; ++r) {
                float d2 = fmaxf(sqm[r] + sqn - 2.0f * c[g][r], 1e-12f);
                panel[(size_t)(r + half * 8) * NROWS + n] = sqrtf(d2);
            }
        }
    }
    __syncthreads();

    // ---- Phase 2: one wave per row (16 waves, 16 rows)
    {
        const int m   = wave;                    // local row 0..15
        const int i   = tm * PANEL_ROWS + m;     // global row
        const int tgt = targets[i];
        const float* prow = panel + (size_t)m * NROWS;

        // positives: the other members of i's group of MM (validated via targets)
        float p0, p1, p2;
        {
            float pv = 1e30f;
            if (lane < MM) {
                int j = (i & ~(MM - 1)) + lane;
                if (j != i && targets[j] == tgt) pv = prow[j];
            }
            float q0 = __shfl(pv, 0, 32), q1 = __shfl(pv, 1, 32);
            float q2 = __shfl(pv, 2, 32), q3 = __shfl(pv, 3, 32);
            float t;
            if (q0 > q1) { t = q0; q0 = q1; q1 = t; }
            if (q2 > q3) { t = q2; q2 = q3; q3 = t; }
            if (q0 > q2) { t = q0; q0 = q2; q2 = t; }
            if (q1 > q3) { t = q1; q1 = q3; q3 = t; }
            if (q1 > q2) { t = q1; q1 = q2; q2 = t; }
            p0 = q0; p1 = q1; p2 = q2;           // 3 smallest (4th is the INF self slot)
        }

        // pass 1: negative sum / sumsq (mean & std are sort-invariant)
        float sum = 0.0f, sumsq = 0.0f;
        for (int j = lane; j < NROWS; j += 32) {
            if (targets[j] == tgt) continue;
            float d = prow[j];
            sum += d; sumsq += d * d;
        }
#pragma unroll
        for (int off = 16; off > 0; off >>= 1) {
            sum   += __shfl_xor(sum, off, 32);
            sumsq += __shfl_xor(sumsq, off, 32);
        }
        const float nneg    = (float)NNEG;
        const float mean    = sum / nneg;
        const float var     = fmaxf(sumsq / nneg - mean * mean, 1e-20f);
        const float inv2var = 0.5f / var;

        // pass 2: per-lane top-3 of (logprob + gumbel)
        float ts[3] = {-1e30f, -1e30f, -1e30f};
        float tv[3] = {0.0f, 0.0f, 0.0f};
        for (int j = lane; j < NROWS; j += 32) {
            if (targets[j] == tgt) continue;
            float d  = prow[j];
            float dm = d - mean;
            top3_insert(dm * dm * inv2var + gumbelf(i, j), d, ts, tv);
        }

        // 3 extraction rounds of wave-wide argmax (descending score order)
        float samp[KPOS];
        int taken = 0;
#pragma unroll
        for (int r = 0; r < KPOS; ++r) {
            float s = (taken < 3) ? ts[taken] : -1e30f;
            float v = (taken < 3) ? tv[taken] : 0.0f;
            int   idx = lane;
#pragma unroll
            for (int off = 16; off > 0; off >>= 1) {
                float s2 = __shfl_xor(s, off, 32);
                float v2 = __shfl_xor(v, off, 32);
                int   i2 = __shfl_xor(idx, off, 32);
                if (s2 > s || (s2 == s && i2 < idx)) { s = s2; v = v2; idx = i2; }
            }
            samp[r] = v;
            if (lane == idx) taken++;
        }

        // finalize row on lane 0
        if (lane == 0) {
            const float thresh = p2 + 0.05f;
            float cnt = 0.0f, neg_terms = 0.0f, neg0 = 0.0f;
            bool  any = false, have0 = false;
#pragma unroll
            for (int r = 0; r < KPOS; ++r) {
                const float sd = samp[r];
                if (sd < thresh) {
                    cnt += 1.0f;
                    neg_terms += softplusf(20.0f * (1.0f - sd));
                    if (!have0) { neg0 = sd; have0 = true; }
                    any = true;
                }
            }
            const float pos_loss = (0.5f / 3.0f) *
                (softplusf(-2.0f * (1.0f - p0)) +
                 softplusf(-2.0f * (1.0f - p1)) +
                 softplusf(-2.0f * (1.0f - p2)));
            const float neg_loss = 0.05f * neg_terms / fmaxf(cnt, 1.0f);
            const float row_loss = any ? (pos_loss + neg_loss) : 0.0f;
            const float err = (any && (p0 < neg0 - 0.1f)) ? 1.0f : 0.0f;

            atomicAdd(&accum[0], row_loss);
            atomicAdd(&accum[1], err);
            atomicAdd(&accum[2], p0 + p1 + p2);
            atomicAdd(&accum[3], sum);
        }
    }
}

__global__ void finalize_kernel(const float* __restrict__ accum, float* __restrict__ out) {
    if (threadIdx.x == 0) {
        out[0] = accum[0] / (float)NROWS;                        // loss
        out[1] = 1.0f - accum[1] / (float)NROWS;                 // prec
        out[2] = accum[2] / ((float)NROWS * (float)KPOS);        // pos_d
        out[3] = accum[3] / ((float)NROWS * (float)NNEG);        // neg_d
    }
}

// ---------------------------------------------------------------- launch

extern "C" void kernel_launch(void* const* d_in, const int* in_sizes, int n_in,
                              void* d_out, int out_size, void* d_ws, size_t ws_size,
                              hipStream_t stream) {
    const float* A       = (const float*)d_in[0];   // (4096,128) f32
    const int*   targets = (const int*)d_in[1];     // (4096,)    i32
    float*       out     = (float*)d_out;           // 4 scalars

    float* sq    = (float*)d_ws;                    // 4096 f32
    float* accum = sq + NROWS;                      // 4 f32

    // Allow the 256 KB dynamic-LDS panel (WGP LDS limit is 320 KB on gfx1250).
    (void)hipFuncSetAttribute((const void*)panel_loss_kernel,
                              hipFuncAttributeMaxDynamicSharedMemorySize,
                              (int)PANEL_BYTES);

    init_accum_kernel<<<1, 32, 0, stream>>>(accum);
    sqnorm_kernel<<<NROWS / 8, 256, 0, stream>>>(A, sq);
    panel_loss_kernel<<<NROWS / PANEL_ROWS, 32 * NWAVES, PANEL_BYTES, stream>>>(A, sq, targets, accum);
    finalize_kernel<<<1, 32, 0, stream>>>(accum, out);
}